// HetGTAN_76682346102816
// MI455X (gfx1250) — compile-verified
//
#include <hip/hip_runtime.h>
#include <hip/hip_bf16.h>

typedef float v2f __attribute__((ext_vector_type(2)));
typedef float v8f __attribute__((ext_vector_type(8)));

#define HID 64
#define SEMD 128
#define HOPS 5

__device__ __forceinline__ float leaky02(float v) { return v > 0.f ? v : 0.2f * v; }

__device__ __forceinline__ float wave_sum(float v) {
#pragma unroll
  for (int off = 16; off; off >>= 1) v += __shfl_xor(v, off, 32);
  return v;
}

// ---------------------------------------------------------------------------
// Generic GEMM: C[M, NTC*16] = act(A[M,K] @ B[K, NTC*16] + bias)
// One wave32 per 16-row tile, V_WMMA_F32_16X16X4_F32, K handled with tail mask.
// A frag (16x4 f32, 2 VGPR): lanes 0-15 row M=lane K={k,k+1}; lanes 16-31 K={k+2,k+3}.
// B frag (4x16 f32, 2 VGPR): lanes 0-15 N=lane K rows {k,k+1}; lanes 16-31 {k+2,k+3}.
// C frag (8 VGPR): vgpr r -> M = r (lanes 0-15) / r+8 (lanes 16-31), N = lane%16.
// ---------------------------------------------------------------------------
template <int NTC>
__global__ void gemm_bias_act_wmma(const float* __restrict__ A, const float* __restrict__ B,
                                   const float* __restrict__ bias, float* __restrict__ C,
                                   int M, int K, int act) {
  const int N = NTC * 16;
  const int wave = (blockIdx.x * blockDim.x + threadIdx.x) >> 5;
  const int lane = threadIdx.x & 31;
  const int m0 = wave << 4;
  if (m0 >= M) return;
  const int half = lane >> 4;
  const int l16 = lane & 15;
  v8f acc[NTC] = {};
  int row = m0 + l16;
  if (row >= M) row = M - 1;  // clamp (loads only; stores guarded)
  for (int k = 0; k < K; k += 4) {
    const int ka = k + half * 2;
    v2f a;
    a.x = (ka < K) ? A[(size_t)row * K + ka] : 0.f;
    a.y = (ka + 1 < K) ? A[(size_t)row * K + ka + 1] : 0.f;
#pragma unroll
    for (int nt = 0; nt < NTC; ++nt) {
      const int col = (nt << 4) + l16;
      v2f b;
      b.x = (ka < K) ? B[(size_t)ka * N + col] : 0.f;
      b.y = (ka + 1 < K) ? B[(size_t)(ka + 1) * N + col] : 0.f;
      acc[nt] = __builtin_amdgcn_wmma_f32_16x16x4_f32(false, a, false, b, (short)0,
                                                      acc[nt], false, false);
    }
  }
#pragma unroll
  for (int nt = 0; nt < NTC; ++nt) {
    const int col = (nt << 4) + l16;
    const float bv = bias[col];
#pragma unroll
    for (int r = 0; r < 8; ++r) {
      const int rr = m0 + (half << 3) + r;
      if (rr < M) {
        float v = acc[nt][r] + bv;
        if (act) v = fmaxf(v, 0.f);
        C[(size_t)rr * N + col] = v;
      }
    }
  }
}

// ---------------------------------------------------------------------------
// Semantic attention score, fused: acc_out += sum_rows( tanh(Z[16,64]@W + b) @ q )
// WMMA tile compute + full-tile reduction -> single atomicAdd per wave.
// ---------------------------------------------------------------------------
__global__ void sem_score_wmma(const float* __restrict__ Z, const float* __restrict__ W,
                               const float* __restrict__ bias, const float* __restrict__ q,
                               float* __restrict__ acc_out, int M) {
  const int wave = (blockIdx.x * blockDim.x + threadIdx.x) >> 5;
  const int lane = threadIdx.x & 31;
  const int m0 = wave << 4;
  if (m0 >= M) return;
  const int half = lane >> 4;
  const int l16 = lane & 15;
  v8f acc[SEMD / 16] = {};
  const int row = m0 + l16;
  for (int k = 0; k < HID; k += 4) {
    const int ka = k + half * 2;
    v2f a;
    a.x = Z[(size_t)row * HID + ka];
    a.y = Z[(size_t)row * HID + ka + 1];
#pragma unroll
    for (int nt = 0; nt < SEMD / 16; ++nt) {
      const int col = (nt << 4) + l16;
      v2f b;
      b.x = W[(size_t)ka * SEMD + col];
      b.y = W[(size_t)(ka + 1) * SEMD + col];
      acc[nt] = __builtin_amdgcn_wmma_f32_16x16x4_f32(false, a, false, b, (short)0,
                                                      acc[nt], false, false);
    }
  }
  float s = 0.f;
#pragma unroll
  for (int nt = 0; nt < SEMD / 16; ++nt) {
    const int col = (nt << 4) + l16;
    const float bv = bias[col];
    const float qv = q[col];
#pragma unroll
    for (int r = 0; r < 8; ++r) s += tanhf(acc[nt][r] + bv) * qv;
  }
  s = wave_sum(s);
  if (lane == 0) atomicAdd(acc_out, s);
}

// ---------------------------------------------------------------------------
// CSR build
// ---------------------------------------------------------------------------
__global__ void csr_count(const int* __restrict__ src, int* __restrict__ counts, int E) {
  int e = blockIdx.x * blockDim.x + threadIdx.x;
  if (e < E) atomicAdd(&counts[src[e]], 1);
}

__global__ void csr_scan(const int* __restrict__ counts, int* __restrict__ offsets, int n) {
  __shared__ int part[1024];
  const int tid = threadIdx.x;
  const int nth = blockDim.x;
  const int chunk = (n + nth - 1) / nth;
  const int s0 = tid * chunk;
  const int s1 = (s0 + chunk < n) ? s0 + chunk : n;
  int s = 0;
  for (int i = s0; i < s1; ++i) s += counts[i];
  part[tid] = s;
  __syncthreads();
  if (tid == 0) {
    int run = 0;
    for (int i = 0; i < nth; ++i) {
      int t = part[i];
      part[i] = run;
      run += t;
    }
    offsets[n] = run;
  }
  __syncthreads();
  int run = part[tid];
  for (int i = s0; i < s1; ++i) {
    offsets[i] = run;
    run += counts[i];
  }
}

__global__ void csr_fill(const int* __restrict__ src, const int* __restrict__ tgt,
                         int* __restrict__ cursor, int* __restrict__ adj, int E) {
  int e = blockIdx.x * blockDim.x + threadIdx.x;
  if (e < E) {
    int pos = atomicAdd(&cursor[src[e]], 1);
    adj[pos] = tgt[e];
  }
}

// ---------------------------------------------------------------------------
// h1[j] = dot(h[j,:], a[:])  (one wave per node, float2 per lane -> b64 loads)
// ---------------------------------------------------------------------------
__global__ void dot_hid(const float* __restrict__ h, const float* __restrict__ a,
                        float* __restrict__ out, int n) {
  const int wave = (blockIdx.x * blockDim.x + threadIdx.x) >> 5;
  const int lane = threadIdx.x & 31;
  if (wave >= n) return;
  const float2 hv = *(const float2*)&h[(size_t)wave * HID + 2 * lane];
  const float2 av = *(const float2*)&a[2 * lane];
  float v = hv.x * av.x + hv.y * av.y;
  v = wave_sum(v);
  if (lane == 0) out[wave] = v;
}

// ---------------------------------------------------------------------------
// GAT-style edge aggregation, gather form (CSR keyed by source node):
// out[i] = (w2*x[i] + sum_j w1_j * h[t_j]) / (w2 + sum_j w1_j)
// w2 = exp(leaky(x[i]·a1 + x[i]·a2)), w1_j = exp(leaky(x[i]·a1 + h1t[t_j]))
// One wave per node; lane owns dims {2*lane, 2*lane+1} -> one b64 gather per
// h-row per lane; L2-resident random gathers, zero atomics in hot loop.
// ---------------------------------------------------------------------------
__global__ void gather_agg(const float* __restrict__ x, const float* __restrict__ h,
                           const float* __restrict__ h1t, const int* __restrict__ offsets,
                           const int* __restrict__ adj, const float* __restrict__ a1,
                           const float* __restrict__ a2, float* __restrict__ out, int n) {
  const int wave = (blockIdx.x * blockDim.x + threadIdx.x) >> 5;
  const int lane = threadIdx.x & 31;
  if (wave >= n) return;
  const float2 xv = *(const float2*)&x[(size_t)wave * HID + 2 * lane];
  const float2 a1v = *(const float2*)&a1[2 * lane];
  const float2 a2v = *(const float2*)&a2[2 * lane];
  float p1 = xv.x * a1v.x + xv.y * a1v.y;
  float p2 = xv.x * a2v.x + xv.y * a2v.y;
  p1 = wave_sum(p1);
  p2 = wave_sum(p2);
  const float x1 = p1;
  const float w2 = expf(leaky02(x1 + p2));
  float acc0 = w2 * xv.x, acc1 = w2 * xv.y, div = w2;
  const int beg = offsets[wave];
  const int end = offsets[wave + 1];
  for (int j = beg; j < end; ++j) {
    const int t = adj[j];
    const float w1 = expf(leaky02(x1 + h1t[t]));
    const float2 hv = *(const float2*)&h[(size_t)t * HID + 2 * lane];
    acc0 += w1 * hv.x;
    acc1 += w1 * hv.y;
    div += w1;
  }
  const float inv = 1.f / div;
  float2 o;
  o.x = acc0 * inv;
  o.y = acc1 * inv;
  *(float2*)&out[(size_t)wave * HID + 2 * lane] = o;
}

// ---------------------------------------------------------------------------
// beta = softmax(acc/N); h = elu(beta0*e0 + beta1*e1)
// ---------------------------------------------------------------------------
__global__ void sem_combine(const float* __restrict__ e0, const float* __restrict__ e1,
                            const float* __restrict__ acc, float* __restrict__ hout,
                            int total, float invN) {
  int idx = blockIdx.x * blockDim.x + threadIdx.x;
  if (idx >= total) return;
  const float s0 = acc[0] * invN, s1 = acc[1] * invN;
  const float m = fmaxf(s0, s1);
  const float x0 = expf(s0 - m), x1 = expf(s1 - m);
  const float inv = 1.f / (x0 + x1);
  const float v = (x0 * inv) * e0[idx] + (x1 * inv) * e1[idx];
  hout[idx] = v > 0.f ? v : (expf(v) - 1.f);
}

// ---------------------------------------------------------------------------
// out[n,0:4] = h[n,:] @ W2[64,4] + b2   (float4 / b128 loads)
// ---------------------------------------------------------------------------
__global__ void final_proj(const float* __restrict__ h, const float* __restrict__ W2,
                           const float* __restrict__ b2, float* __restrict__ out, int n) {
  int i = blockIdx.x * blockDim.x + threadIdx.x;
  if (i >= n) return;
  float a0 = b2[0], a1 = b2[1], a2 = b2[2], a3 = b2[3];
#pragma unroll 4
  for (int d = 0; d < HID; d += 4) {
    const float4 hv = *(const float4*)&h[(size_t)i * HID + d];
    const float4 r0 = *(const float4*)&W2[(d + 0) * 4];
    const float4 r1 = *(const float4*)&W2[(d + 1) * 4];
    const float4 r2 = *(const float4*)&W2[(d + 2) * 4];
    const float4 r3 = *(const float4*)&W2[(d + 3) * 4];
    a0 += hv.x * r0.x + hv.y * r1.x + hv.z * r2.x + hv.w * r3.x;
    a1 += hv.x * r0.y + hv.y * r1.y + hv.z * r2.y + hv.w * r3.y;
    a2 += hv.x * r0.z + hv.y * r1.z + hv.z * r2.z + hv.w * r3.z;
    a3 += hv.x * r0.w + hv.y * r1.w + hv.z * r2.w + hv.w * r3.w;
  }
  float4 o;
  o.x = a0; o.y = a1; o.z = a2; o.w = a3;
  *(float4*)&out[(size_t)i * 4] = o;
}

// ---------------------------------------------------------------------------

static inline char* bump(char*& p, size_t bytes) {
  char* r = p;
  p += (bytes + 255) & ~(size_t)255;
  return r;
}

extern "C" void kernel_launch(void* const* d_in, const int* in_sizes, int n_in,
                              void* d_out, int out_size, void* d_ws, size_t ws_size,
                              hipStream_t stream) {
  (void)n_in; (void)out_size; (void)ws_size;
  const float* x_author = (const float*)d_in[0];
  const float* x_paper  = (const float*)d_in[1];
  const float* W1_a = (const float*)d_in[2];
  const float* b1_a = (const float*)d_in[3];
  const float* W1_p = (const float*)d_in[4];
  const float* b1_p = (const float*)d_in[5];
  const float* a1_ap = (const float*)d_in[6];
  const float* a2_ap = (const float*)d_in[7];
  const float* a1_aa = (const float*)d_in[8];
  const float* a2_aa = (const float*)d_in[9];
  const float* a1_pa = (const float*)d_in[10];
  const float* a2_pa = (const float*)d_in[11];
  const float* a1_pp = (const float*)d_in[12];
  const float* a2_pp = (const float*)d_in[13];
  const float* semW_a = (const float*)d_in[14];
  const float* semb_a = (const float*)d_in[15];
  const float* semq_a = (const float*)d_in[16];
  const float* semW_p = (const float*)d_in[17];
  const float* semb_p = (const float*)d_in[18];
  const float* semq_p = (const float*)d_in[19];
  const float* W2 = (const float*)d_in[20];
  const float* b2 = (const float*)d_in[21];
  const int* ei_ap = (const int*)d_in[22];
  const int* ei_aa = (const int*)d_in[23];
  const int* ei_pa = (const int*)d_in[24];
  const int* ei_pp = (const int*)d_in[25];

  const int Ka = in_sizes[2] / HID;        // 334
  const int Kp = in_sizes[4] / HID;        // 512
  const int NA = in_sizes[0] / Ka;         // 40000
  const int NP = in_sizes[1] / Kp;         // 40000
  const int E = in_sizes[22] / 2;          // 800000
  const size_t featA = (size_t)NA * HID * sizeof(float);
  const size_t featP = (size_t)NP * HID * sizeof(float);
  const int NMAX = NA > NP ? NA : NP;

  // ---- workspace layout (bump allocation, deterministic) ----
  char* p = (char*)d_ws;
  float* xA = (float*)bump(p, featA);
  float* xP = (float*)bump(p, featP);
  float* hA[2] = {(float*)bump(p, featA), (float*)bump(p, featA)};
  float* hP[2] = {(float*)bump(p, featP), (float*)bump(p, featP)};
  float* eA0 = (float*)bump(p, featA);
  float* eA1 = (float*)bump(p, featA);
  float* eP0 = (float*)bump(p, featP);
  float* eP1 = (float*)bump(p, featP);
  float* h1ap = (float*)bump(p, (size_t)NP * sizeof(float));
  float* h1aa = (float*)bump(p, (size_t)NA * sizeof(float));
  float* h1pa = (float*)bump(p, (size_t)NA * sizeof(float));
  float* h1pp = (float*)bump(p, (size_t)NP * sizeof(float));
  float* acc = (float*)bump(p, 4 * sizeof(float));  // [A0,A1,P0,P1]
  int* off_ap = (int*)bump(p, (size_t)(NA + 1) * sizeof(int));
  int* off_aa = (int*)bump(p, (size_t)(NA + 1) * sizeof(int));
  int* off_pa = (int*)bump(p, (size_t)(NP + 1) * sizeof(int));
  int* off_pp = (int*)bump(p, (size_t)(NP + 1) * sizeof(int));
  int* adj_ap = (int*)bump(p, (size_t)E * sizeof(int));
  int* adj_aa = (int*)bump(p, (size_t)E * sizeof(int));
  int* adj_pa = (int*)bump(p, (size_t)E * sizeof(int));
  int* adj_pp = (int*)bump(p, (size_t)E * sizeof(int));
  int* counts = (int*)bump(p, (size_t)NMAX * sizeof(int));
  int* cursor = (int*)bump(p, (size_t)NMAX * sizeof(int));

  const int eBlocks = (E + 255) / 256;
  auto wgrid = [](int waves) { return (waves + 7) / 8; };  // 256 thr = 8 wave32

  // ---- CSR build (once per call, reused across 5 hops) ----
  struct Rel { const int* ei; int ns; int* off; int* adj; };
  const Rel rels[4] = {{ei_ap, NA, off_ap, adj_ap},
                       {ei_aa, NA, off_aa, adj_aa},
                       {ei_pa, NP, off_pa, adj_pa},
                       {ei_pp, NP, off_pp, adj_pp}};
  for (int r = 0; r < 4; ++r) {
    hipMemsetAsync(counts, 0, (size_t)rels[r].ns * sizeof(int), stream);
    csr_count<<<eBlocks, 256, 0, stream>>>(rels[r].ei, counts, E);
    csr_scan<<<1, 1024, 0, stream>>>(counts, rels[r].off, rels[r].ns);
    hipMemcpyAsync(cursor, rels[r].off, (size_t)rels[r].ns * sizeof(int),
                   hipMemcpyDeviceToDevice, stream);
    csr_fill<<<eBlocks, 256, 0, stream>>>(rels[r].ei, rels[r].ei + E, cursor, rels[r].adj, E);
  }

  // ---- input projections (WMMA f32 GEMM + bias + ReLU) ----
  const int tilesA = (NA + 15) / 16;
  const int tilesP = (NP + 15) / 16;
  gemm_bias_act_wmma<HID / 16><<<wgrid(tilesA), 256, 0, stream>>>(x_author, W1_a, b1_a, xA, NA, Ka, 1);
  gemm_bias_act_wmma<HID / 16><<<wgrid(tilesP), 256, 0, stream>>>(x_paper, W1_p, b1_p, xP, NP, Kp, 1);
  hipMemcpyAsync(hA[0], xA, featA, hipMemcpyDeviceToDevice, stream);
  hipMemcpyAsync(hP[0], xP, featP, hipMemcpyDeviceToDevice, stream);

  // ---- hop loop ----
  int cur = 0;
  for (int i = 0; i < HOPS; ++i) {
    const float* a2ap = a2_ap + i * HID;
    const float* a2aa = a2_aa + i * HID;
    const float* a2pa = a2_pa + i * HID;
    const float* a2pp = a2_pp + i * HID;
    // target-side attention scalars h1 = h @ a2
    dot_hid<<<wgrid(NP), 256, 0, stream>>>(hP[cur], a2ap, h1ap, NP);
    dot_hid<<<wgrid(NA), 256, 0, stream>>>(hA[cur], a2aa, h1aa, NA);
    dot_hid<<<wgrid(NA), 256, 0, stream>>>(hA[cur], a2pa, h1pa, NA);
    dot_hid<<<wgrid(NP), 256, 0, stream>>>(hP[cur], a2pp, h1pp, NP);
    // edge aggregation (gather, L2-resident)
    gather_agg<<<wgrid(NA), 256, 0, stream>>>(xA, hP[cur], h1ap, off_ap, adj_ap,
                                              a1_ap + i * HID, a2ap, eA0, NA);
    gather_agg<<<wgrid(NA), 256, 0, stream>>>(xA, hA[cur], h1aa, off_aa, adj_aa,
                                              a1_aa + i * HID, a2aa, eA1, NA);
    gather_agg<<<wgrid(NP), 256, 0, stream>>>(xP, hA[cur], h1pa, off_pa, adj_pa,
                                              a1_pa + i * HID, a2pa, eP0, NP);
    gather_agg<<<wgrid(NP), 256, 0, stream>>>(xP, hP[cur], h1pp, off_pp, adj_pp,
                                              a1_pp + i * HID, a2pp, eP1, NP);
    // semantic attention scores (WMMA + fused global reduction)
    hipMemsetAsync(acc, 0, 4 * sizeof(float), stream);
    sem_score_wmma<<<wgrid(tilesA), 256, 0, stream>>>(eA0, semW_a + i * HID * SEMD,
                                                      semb_a + i * SEMD, semq_a + i * SEMD,
                                                      acc + 0, NA);
    sem_score_wmma<<<wgrid(tilesA), 256, 0, stream>>>(eA1, semW_a + i * HID * SEMD,
                                                      semb_a + i * SEMD, semq_a + i * SEMD,
                                                      acc + 1, NA);
    sem_score_wmma<<<wgrid(tilesP), 256, 0, stream>>>(eP0, semW_p + i * HID * SEMD,
                                                      semb_p + i * SEMD, semq_p + i * SEMD,
                                                      acc + 2, NP);
    sem_score_wmma<<<wgrid(tilesP), 256, 0, stream>>>(eP1, semW_p + i * HID * SEMD,
                                                      semb_p + i * SEMD, semq_p + i * SEMD,
                                                      acc + 3, NP);
    // beta softmax + weighted combine + ELU -> next h
    sem_combine<<<(NA * HID + 255) / 256, 256, 0, stream>>>(eA0, eA1, acc + 0, hA[1 - cur],
                                                            NA * HID, 1.f / (float)NA);
    sem_combine<<<(NP * HID + 255) / 256, 256, 0, stream>>>(eP0, eP1, acc + 2, hP[1 - cur],
                                                            NP * HID, 1.f / (float)NP);
    cur = 1 - cur;
  }

  // ---- output head (author nodes) ----
  final_proj<<<(NA + 255) / 256, 256, 0, stream>>>(hA[cur], W2, b2, (float*)d_out, NA);
}